// GINEncoder_12635793785089
// MI455X (gfx1250) — compile-verified
//
#include <hip/hip_runtime.h>
#include <stdint.h>

#define N_NODES  100000
#define N_EDGES  1600000
#define DIM      128
#define N_LAYERS 3
#define N_GRAPHS 2048
#define M_TILE   64

typedef __attribute__((ext_vector_type(2))) float    v2f;
typedef __attribute__((ext_vector_type(8))) float    v8f;
typedef __attribute__((ext_vector_type(4))) unsigned v4u;
typedef __attribute__((ext_vector_type(8))) int      v8i;
typedef __attribute__((ext_vector_type(4))) int      v4i;

// ---------------------------------------------------------------------------
// TDM: DMA one 128x128 f32 matrix (64KB) from global into LDS.
// D# group0: [1:0]=count=1, [63:32]=lds byte addr, [120:64]=global byte addr,
//            [127:126]=type=2.  Group1: data_size=4B, tensor 128x128,
//            tile 128x128, dim0 stride=128.  Groups 2/3 unused (2D tensor).
// Issue from a single wave (TDM ignores EXEC; operands are wave-uniform).
// Completion tracked with TENSORcnt.
// ---------------------------------------------------------------------------
__device__ __forceinline__ void tdm_load_w128(const float* Wg, float* ldsDst) {
    const uint64_t ga = (uint64_t)(uintptr_t)Wg;
    // generic pointers to LDS carry the byte offset in the low 32 bits
    const uint32_t la = (uint32_t)(uintptr_t)ldsDst;
    v4u g0;
    g0.x = 1u;                                               // count=1
    g0.y = la;                                               // lds_addr
    g0.z = (uint32_t)ga;                                     // global_addr[31:0]
    g0.w = (uint32_t)((ga >> 32) & 0x1FFFFFFu) | (2u << 30); // addr[56:32]|type=2
    v8i g1;
    g1[0] = 0x00020000;        // workgroup_mask=0, data_size=2 (4 bytes)
    g1[1] = (int)(128u << 16); // tensor_dim0[15:0]=128  (bits 63:48)
    g1[2] = (int)(128u << 16); // tensor_dim0[31:16]=0, tensor_dim1[15:0]=128
    g1[3] = (int)(128u << 16); // tensor_dim1[31:16]=0, tile_dim0=128
    g1[4] = 128;               // tile_dim1=128, tile_dim2=0
    g1[5] = 128;               // tensor_dim0_stride[31:0]=128
    g1[6] = 0;                 // stride[47:32]=0, tensor_dim1_stride lo=0
    g1[7] = 0;
    const v4i z4 = {0, 0, 0, 0};
    const v8i z8 = {0, 0, 0, 0, 0, 0, 0, 0};
    __builtin_amdgcn_tensor_load_to_lds(g0, g1, z4, z4, z8, 0);
}

// ---------------------------------------------------------------------------
// Edge scatter-add: agg[dst] += x[src].  One wave32 per edge; each lane moves
// 4 consecutive floats (32 lanes * 4 = 128 = DIM).  Gather is one coalesced
// 512B row per wave; x fits in the 192MB L2 so this runs at L2 rate.
// ---------------------------------------------------------------------------
__global__ __launch_bounds__(256) void gin_scatter(const float* __restrict__ x,
                                                   const int* __restrict__ src,
                                                   const int* __restrict__ dst,
                                                   float* __restrict__ agg) {
    const int gid  = blockIdx.x * blockDim.x + threadIdx.x;
    const int edge = gid >> 5;
    const int lane = gid & 31;
    if (edge >= N_EDGES) return;
    const int s = src[edge];
    const int d = dst[edge];
    const float4 v = *reinterpret_cast<const float4*>(x + (size_t)s * DIM + lane * 4);
    float* out = agg + (size_t)d * DIM + lane * 4;
    atomicAdd(out + 0, v.x);
    atomicAdd(out + 1, v.y);
    atomicAdd(out + 2, v.z);
    atomicAdd(out + 3, v.w);
}

// ---------------------------------------------------------------------------
// Fused GIN MLP for a 64-node tile:
//   x[tile] = relu( relu( (agg+x) @ W1 + b1 ) @ W2 + b2 )     (in-place)
// 256 threads = 8 waves; wave (r,c) owns rows r*16..r*16+15, cols c*64..c*64+63
// => 4 v8f WMMA accumulators per wave, K=128 swept in 4-wide steps with
// v_wmma_f32_16x16x4_f32.  Weights staged via TDM (TENSORcnt), H via VALU
// (needs the agg+x fusion and the row guard).
// ---------------------------------------------------------------------------
__global__ __launch_bounds__(256) void gin_mlp(const float* __restrict__ agg,
                                               float* __restrict__ x,
                                               const float* __restrict__ W1,
                                               const float* __restrict__ b1,
                                               const float* __restrict__ W2,
                                               const float* __restrict__ b2) {
    __shared__ float sW[DIM * DIM];    // 64 KB: current weight matrix (K-major)
    __shared__ float sH[M_TILE * DIM]; // 32 KB: input tile, then mid tile

    const int tid      = threadIdx.x;
    const int lane     = tid & 31;
    const int wave     = tid >> 5;
    const int half     = lane >> 4;   // 0: lanes 0-15, 1: lanes 16-31
    const int l16      = lane & 15;
    const int rowBase  = (wave >> 1) * 16;  // 0,16,32,48
    const int colBase  = (wave & 1) * 64;   // 0,64
    const int nodeBase = blockIdx.x * M_TILE;

    // ---- W1 -> LDS via Tensor Data Mover (wave 0 issues, one DMA per WG) ----
    if (wave == 0) tdm_load_w128(W1, sW);

    // ---- stage H = agg + x (manual: needs add + boundary guard) ----
    for (int i = 0; i < 8; ++i) {
        const int f4   = i * 256 + tid; // 0..2047
        const int r    = f4 >> 5;       // tile row (32 float4 per row)
        const int c4   = f4 & 31;
        const int node = nodeBase + r;
        float4 v = make_float4(0.f, 0.f, 0.f, 0.f);
        if (node < N_NODES) {
            const float4 a = *reinterpret_cast<const float4*>(agg + (size_t)node * DIM + c4 * 4);
            const float4 b = *reinterpret_cast<const float4*>(x   + (size_t)node * DIM + c4 * 4);
            v = make_float4(a.x + b.x, a.y + b.y, a.z + b.z, a.w + b.w);
        }
        *reinterpret_cast<float4*>(sH + f4 * 4) = v;
    }
    if (wave == 0) __builtin_amdgcn_s_wait_tensorcnt(0);
    __syncthreads();

    const v8f vzero = {0.f, 0.f, 0.f, 0.f, 0.f, 0.f, 0.f, 0.f};
    v8f acc[4];

    // ---- GEMM1 ----
    for (int f = 0; f < 4; ++f) acc[f] = vzero;
    for (int k0 = 0; k0 < DIM; k0 += 4) {
        // A frag (16x4 f32): lanes 0-15 rows, VGPR0={K+0|K+2}, VGPR1={K+1|K+3}
        const float* ha = sH + (rowBase + l16) * DIM + k0 + 2 * half;
        v2f a; a.x = ha[0]; a.y = ha[1];
        for (int f = 0; f < 4; ++f) {
            const int n0 = colBase + f * 16;
            const float* wb = sW + (k0 + 2 * half) * DIM + n0 + l16;
            v2f b; b.x = wb[0]; b.y = wb[DIM];
            acc[f] = __builtin_amdgcn_wmma_f32_16x16x4_f32(
                false, a, false, b, (short)0, acc[f], false, false);
        }
    }
    __syncthreads();  // all reads of sH/sW done

    // ---- W2 -> LDS via TDM, overlapped with mid-tile writeback ----
    if (wave == 0) tdm_load_w128(W2, sW);

    // ---- bias + relu, write mid into sH ----
    for (int f = 0; f < 4; ++f) {
        const int n = colBase + f * 16 + l16;
        const float bias = b1[n];
        for (int r = 0; r < 8; ++r) {
            float v = acc[f][r] + bias;
            v = v > 0.f ? v : 0.f;
            sH[(rowBase + r + 8 * half) * DIM + n] = v;  // C/D: VGPR r -> M=r / r+8
        }
    }
    if (wave == 0) __builtin_amdgcn_s_wait_tensorcnt(0);
    __syncthreads();

    // ---- GEMM2 ----
    for (int f = 0; f < 4; ++f) acc[f] = vzero;
    for (int k0 = 0; k0 < DIM; k0 += 4) {
        const float* ha = sH + (rowBase + l16) * DIM + k0 + 2 * half;
        v2f a; a.x = ha[0]; a.y = ha[1];
        for (int f = 0; f < 4; ++f) {
            const int n0 = colBase + f * 16;
            const float* wb = sW + (k0 + 2 * half) * DIM + n0 + l16;
            v2f b; b.x = wb[0]; b.y = wb[DIM];
            acc[f] = __builtin_amdgcn_wmma_f32_16x16x4_f32(
                false, a, false, b, (short)0, acc[f], false, false);
        }
    }

    // ---- bias + relu, store in place ----
    for (int f = 0; f < 4; ++f) {
        const int n = colBase + f * 16 + l16;
        const float bias = b2[n];
        for (int r = 0; r < 8; ++r) {
            const int node = nodeBase + rowBase + r + 8 * half;
            if (node < N_NODES) {
                float v = acc[f][r] + bias;
                v = v > 0.f ? v : 0.f;
                x[(size_t)node * DIM + n] = v;
            }
        }
    }
}

// ---------------------------------------------------------------------------
// Mean pool: wave32 per node scatter-adds its row into sums[batch[node]].
// ---------------------------------------------------------------------------
__global__ __launch_bounds__(256) void gin_pool_scatter(const float* __restrict__ x,
                                                        const int* __restrict__ batch,
                                                        float* __restrict__ sums,
                                                        float* __restrict__ counts) {
    const int gid  = blockIdx.x * blockDim.x + threadIdx.x;
    const int node = gid >> 5;
    const int lane = gid & 31;
    if (node >= N_NODES) return;
    const int g = batch[node];
    const float4 v = *reinterpret_cast<const float4*>(x + (size_t)node * DIM + lane * 4);
    float* out = sums + (size_t)g * DIM + lane * 4;
    atomicAdd(out + 0, v.x);
    atomicAdd(out + 1, v.y);
    atomicAdd(out + 2, v.z);
    atomicAdd(out + 3, v.w);
    if (lane == 0) atomicAdd(counts + g, 1.0f);
}

__global__ __launch_bounds__(256) void gin_pool_final(const float* __restrict__ sums,
                                                      const float* __restrict__ counts,
                                                      float* __restrict__ out) {
    const int i = blockIdx.x * blockDim.x + threadIdx.x;
    if (i >= N_GRAPHS * DIM) return;
    const int g = i >> 7;  // / DIM
    const float c = counts[g];
    out[i] = sums[i] / (c > 1.0f ? c : 1.0f);
}

// ---------------------------------------------------------------------------
extern "C" void kernel_launch(void* const* d_in, const int* in_sizes, int n_in,
                              void* d_out, int out_size, void* d_ws, size_t ws_size,
                              hipStream_t stream) {
    const float* x_in       = (const float*)d_in[0];
    const int*   edge_index = (const int*)d_in[1];
    const int*   batch      = (const int*)d_in[2];
    const float* W1         = (const float*)d_in[3];
    const float* b1         = (const float*)d_in[4];
    const float* W2         = (const float*)d_in[5];
    const float* b2         = (const float*)d_in[6];
    float*       out        = (float*)d_out;

    const size_t feat = (size_t)N_NODES * DIM;
    float* x_cur  = (float*)d_ws;                  // 51.2 MB
    float* agg    = x_cur + feat;                  // 51.2 MB
    float* sums   = agg + feat;                    // 1 MB
    float* counts = sums + (size_t)N_GRAPHS * DIM; // 8 KB

    (void)hipMemcpyAsync(x_cur, x_in, feat * sizeof(float), hipMemcpyDeviceToDevice, stream);

    const int* src = edge_index;            // row 0
    const int* dst = edge_index + N_EDGES;  // row 1

    const int scatter_blocks = (int)(((long long)N_EDGES * 32 + 255) / 256);
    const int mlp_blocks     = (N_NODES + M_TILE - 1) / M_TILE;

    for (int l = 0; l < N_LAYERS; ++l) {
        (void)hipMemsetAsync(agg, 0, feat * sizeof(float), stream);
        gin_scatter<<<scatter_blocks, 256, 0, stream>>>(x_cur, src, dst, agg);
        gin_mlp<<<mlp_blocks, 256, 0, stream>>>(agg, x_cur,
            W1 + (size_t)l * DIM * DIM, b1 + (size_t)l * DIM,
            W2 + (size_t)l * DIM * DIM, b2 + (size_t)l * DIM);
    }

    (void)hipMemsetAsync(sums, 0, ((size_t)N_GRAPHS * DIM + N_GRAPHS) * sizeof(float), stream);
    const int pool_blocks = (int)(((long long)N_NODES * 32 + 255) / 256);
    gin_pool_scatter<<<pool_blocks, 256, 0, stream>>>(x_cur, batch, sums, counts);
    gin_pool_final<<<(N_GRAPHS * DIM + 255) / 256, 256, 0, stream>>>(sums, counts, out);
}